// R3ConvSites_78486232367378
// MI455X (gfx1250) — compile-verified
//
#include <hip/hip_runtime.h>
#include <hip/hip_bf16.h>

// Problem constants from the reference
#define NB      32
#define NSITES  65536
#define N_NGB   13
#define DIM     3
#define NG      48

typedef __attribute__((ext_vector_type(2))) float v2f;
typedef __attribute__((ext_vector_type(8))) float v8f;

// ---------------------------------------------------------------------------
// Kernel 1: build zero-padded 16x16 Wmean into workspace.
// Wmean[d,j] = (1/48) * sum_g sum_kk gdiags[g*3+d, kk] * Wp[kk, j]
//   with Wp[g2*3+d2, j] = wtVC[d2, GnnPerms[g2, j]]
// 256 threads, each owns exactly one slot of the padded 16x16 tile.
// ---------------------------------------------------------------------------
__global__ __launch_bounds__(256) void wmean_prep_kernel(
    const float* __restrict__ wtVC,     // (3, 13)
    const float* __restrict__ gdiags,   // (144, 144)
    const int*   __restrict__ GnnPerms, // (48, 13)
    float* __restrict__ wpad)           // (16, 16) padded Wmean
{
    const int t = threadIdx.x;   // 0..255
    const int m = t >> 4;        // padded row  (d)
    const int k = t & 15;        // padded col  (j)
    float val = 0.0f;
    if (m < DIM && k < N_NGB) {
        float acc = 0.0f;
        for (int g2 = 0; g2 < NG; ++g2) {
            const int perm = GnnPerms[g2 * N_NGB + k];
            for (int d2 = 0; d2 < DIM; ++d2) {
                const float w  = wtVC[d2 * N_NGB + perm];
                const int   kk = g2 * DIM + d2;
                float gsum = 0.0f;
                for (int g = 0; g < NG; ++g)
                    gsum += gdiags[(g * DIM + m) * (NG * DIM) + kk];
                acc += gsum * w;
            }
        }
        val = acc * (1.0f / (float)NG);
    }
    wpad[m * 16 + k] = val;      // rows 3..15 and cols 13..15 are zero
}

// ---------------------------------------------------------------------------
// Kernel 2: gather + WMMA contraction.
// One wave32 per 16-site tile; loops over all 32 batches reusing the
// batch-invariant gather indices held in registers.
//
// f32 16x16x4 operand layout (32-bit A 16x4 / B 4x16):
//   lane = (khalf<<4) | n ;  n = M-row of A = N-col of B
//   VGPR v of A holds A[n][Kbase + 2*khalf + v]
//   VGPR v of B holds B[Kbase + 2*khalf + v][n]
// A (padded Wmean) is zero for K>=13 and M>=3, so garbage B rows for
// j=13..15 (index clamped to 0) contribute exactly zero -> branchless.
// ---------------------------------------------------------------------------
__global__ __launch_bounds__(256) void r3conv_wmma_kernel(
    const float* __restrict__ In,       // (32, 65536)
    const int*   __restrict__ NNsites,  // (13, 65536)
    const float* __restrict__ wpad,     // (16, 16) padded Wmean
    float* __restrict__ out)            // (32, 3, 65536)
{
    const int lane  = threadIdx.x & 31;
    const int wave  = threadIdx.x >> 5;              // 8 waves / block
    const int tile  = blockIdx.x * 8 + wave;         // 0..4095
    const int n     = lane & 15;                     // site-in-tile / M-row
    const int koff  = (lane >> 4) << 1;              // 0 or 2
    const int s     = tile * 16 + n;                 // global site

    // A fragments (uniform weights) + batch-invariant gather indices
    v2f a[4];
    int idx0[4], idx1[4];
    #pragma unroll
    for (int i = 0; i < 4; ++i) {
        const int j0 = 4 * i + koff;
        const int j1 = j0 + 1;
        a[i][0] = wpad[n * 16 + j0];
        a[i][1] = wpad[n * 16 + j1];
        const int jc0 = (j0 < N_NGB) ? j0 : 0;       // clamp: A zeros kill it
        const int jc1 = (j1 < N_NGB) ? j1 : 0;
        idx0[i] = NNsites[jc0 * NSITES + s];
        idx1[i] = NNsites[jc1 * NSITES + s];
    }

    for (int b = 0; b < NB; ++b) {
        const float* __restrict__ inb = In + b * NSITES;
        v8f c = {0.f, 0.f, 0.f, 0.f, 0.f, 0.f, 0.f, 0.f};
        #pragma unroll
        for (int i = 0; i < 4; ++i) {
            v2f bb;
            bb[0] = inb[idx0[i]];
            bb[1] = inb[idx1[i]];
            // D = A x B + C, fp32, 16x16x4
            c = __builtin_amdgcn_wmma_f32_16x16x4_f32(
                    /*neg_a=*/false, a[i],
                    /*neg_b=*/false, bb,
                    /*c_mod=*/(short)0, c,
                    /*reuse_a=*/false, /*reuse_b=*/false);
        }
        // Rows M=0..2 live in VGPRs 0..2 of lanes 0..15
        if (lane < 16) {
            float* ob = out + b * (DIM * NSITES);
            ob[0 * NSITES + s] = c[0];
            ob[1 * NSITES + s] = c[1];
            ob[2 * NSITES + s] = c[2];
        }
    }
}

extern "C" void kernel_launch(void* const* d_in, const int* in_sizes, int n_in,
                              void* d_out, int out_size, void* d_ws, size_t ws_size,
                              hipStream_t stream) {
    const float* In       = (const float*)d_in[0];  // (32,1,65536)
    const float* wtVC     = (const float*)d_in[1];  // (3,13)
    const float* gdiags   = (const float*)d_in[2];  // (144,144)
    const int*   GnnPerms = (const int*)  d_in[3];  // (48,13)
    const int*   NNsites  = (const int*)  d_in[4];  // (13,65536)
    float*       out      = (float*)d_out;          // (32,3,65536)
    float*       wpad     = (float*)d_ws;           // 256 floats scratch

    wmean_prep_kernel<<<1, 256, 0, stream>>>(wtVC, gdiags, GnnPerms, wpad);

    const int tiles  = NSITES / 16;      // 4096 waves
    const int blocks = tiles / 8;        // 8 waves per 256-thread block
    r3conv_wmma_kernel<<<blocks, 256, 0, stream>>>(In, NNsites, wpad, out);
}